// SubGNN_Contrastive_8778913153582
// MI455X (gfx1250) — compile-verified
//
#include <hip/hip_runtime.h>
#include <math.h>

#define N_NODES  20000
#define N_EDGES  320000
#define FDIM     64
#define N_RUNS   10
#define N_GRAPHS 128
#define N_CLASSES 10
#define RN       (N_RUNS * N_NODES)          /* 200000 rows */
#define BN_EPS   1e-5f

typedef float v2f __attribute__((ext_vector_type(2)));
typedef float v8f __attribute__((ext_vector_type(8)));

/* ---------------- h0 = dropmask ? 0 : x, replicated over runs ------------- */
__global__ void init_h_kernel(const float* __restrict__ x,
                              const int* __restrict__ mask,
                              float* __restrict__ h) {
    long p = (long)blockIdx.x * 4 + (threadIdx.x >> 6);   // (r,n) pair, 200000 total
    int  c = threadIdx.x & 63;
    int  n = (int)(p % N_NODES);
    float v = mask[p] ? 0.f : x[(long)n * FDIM + c];
    h[p * FDIM + c] = v;
}

/* ---------------- edge scatter-add: agg[r,dst] += h[r,src] ---------------- */
__global__ void scatter_add_kernel(const float* __restrict__ h,
                                   const int* __restrict__ src,
                                   const int* __restrict__ dst,
                                   float* __restrict__ agg) {
    int b   = blockIdx.x;                 // 800000 blocks: 80000 per run, 4 edges each
    int r   = b / (N_EDGES / 4);
    int e   = (b % (N_EDGES / 4)) * 4 + (threadIdx.x >> 6);
    int c   = threadIdx.x & 63;
    long base = (long)r * N_NODES * FDIM;
    float v = h[base + (long)src[e] * FDIM + c];
    atomicAdd(&agg[base + (long)dst[e] * FDIM + c], v);
}

/* ---------------- fused 64x64 GEMM, fp32 WMMA 16x16x4 ---------------------
 * MODE 0: A = in0 + in1                (GIN combine, h + agg)
 * MODE 1: A = relu(in0*scale + shift)  (BN1 affine + ReLU fused on load)
 * out[m,:] = A[m,:] @ W + bias ; safe in-place (row-band local).
 * B staged pair-packed in LDS: Wp[k/2][n] = {W[k][n], W[k+1][n]} so every
 * fragment is one 8B-aligned ds_load_b64 straight into a VGPR pair.        */
template<int MODE>
__global__ void gemm64_kernel(const float* __restrict__ in0,
                              const float* __restrict__ in1,
                              const float* __restrict__ W,
                              const float* __restrict__ bias,
                              const float* __restrict__ scaleshift,
                              float* out) {
    __shared__ v2f   Wp[32 * 64];   // [k-pair][n]
    __shared__ float bl[64];
    __shared__ float sc[64];
    __shared__ float sh[64];
    int tid = threadIdx.x;
    { // stage 16KB weight tile pair-packed + vectors
        float* Wf = (float*)Wp;
        for (int i = tid; i < 4096; i += 128) {
            int k = i >> 6, n = i & 63;
            Wf[((k >> 1) * 64 + n) * 2 + (k & 1)] = W[i];
        }
        if (tid < 64) {
            bl[tid] = bias[tid];
            if (MODE == 1) { sc[tid] = scaleshift[tid]; sh[tid] = scaleshift[64 + tid]; }
        }
    }
    __syncthreads();

    int  wave = tid >> 5, lane = tid & 31;
    long m0   = ((long)blockIdx.x * 4 + wave) * 16;   // 12500 bands, exact
    int  M    = lane & 15;
    int  kk   = (lane >> 4) * 2;

    const float* arow0 = in0 + (m0 + M) * FDIM;
    const float* arow1 = (MODE == 0) ? (in1 + (m0 + M) * FDIM) : nullptr;

    v8f acc0 = {}, acc1 = {}, acc2 = {}, acc3 = {};
#pragma unroll
    for (int k0 = 0; k0 < 64; k0 += 4) {
        int ka = k0 + kk;                    // even
        v2f a;
        if (MODE == 0) {
            a[0] = arow0[ka]     + arow1[ka];
            a[1] = arow0[ka + 1] + arow1[ka + 1];
        } else {
            float x0 = arow0[ka]     * sc[ka]     + sh[ka];
            float x1 = arow0[ka + 1] * sc[ka + 1] + sh[ka + 1];
            a[0] = x0 > 0.f ? x0 : 0.f;
            a[1] = x1 > 0.f ? x1 : 0.f;
        }
        int kbase = (ka >> 1) * 64 + M;
        v2f b0 = Wp[kbase];
        v2f b1 = Wp[kbase + 16];
        v2f b2 = Wp[kbase + 32];
        v2f b3 = Wp[kbase + 48];
        acc0 = __builtin_amdgcn_wmma_f32_16x16x4_f32(false, a, false, b0, (short)0, acc0, false, false);
        acc1 = __builtin_amdgcn_wmma_f32_16x16x4_f32(false, a, false, b1, (short)0, acc1, false, false);
        acc2 = __builtin_amdgcn_wmma_f32_16x16x4_f32(false, a, false, b2, (short)0, acc2, false, false);
        acc3 = __builtin_amdgcn_wmma_f32_16x16x4_f32(false, a, false, b3, (short)0, acc3, false, false);
    }
    // D layout: row = m0 + j + 8*(lane>>4), col = nt*16 + (lane&15)
    int rowoff = (lane >> 4) * 8;
    v8f accs[4] = {acc0, acc1, acc2, acc3};
#pragma unroll
    for (int nt = 0; nt < 4; ++nt) {
        int   col = nt * 16 + M;
        float bv  = bl[col];
#pragma unroll
        for (int j = 0; j < 8; ++j)
            out[(m0 + rowoff + j) * FDIM + col] = accs[nt][j] + bv;
    }
}

/* ---------------- BN column stats: f64 sum / sumsq via atomics ------------ */
__global__ void bn_stats_kernel(const float* __restrict__ t, double* __restrict__ stats) {
    int tid = blockIdx.x * blockDim.x + threadIdx.x;  // 512*256 = 131072, mult of 64
    int col = tid & 63;
    double s = 0.0, s2 = 0.0;
    const long total = (long)RN * FDIM;
    for (long i = tid; i < total; i += 131072) {
        float v = t[i];
        s += (double)v;
        s2 += (double)v * (double)v;
    }
    atomicAdd(&stats[col], s);
    atomicAdd(&stats[64 + col], s2);
}

__global__ void bn_finalize_kernel(const double* __restrict__ stats,
                                   const float* __restrict__ g,
                                   const float* __restrict__ b,
                                   float* __restrict__ scaleshift) {
    int c = threadIdx.x;   // 64
    double mu  = stats[c] / (double)RN;
    double var = stats[64 + c] / (double)RN - mu * mu;
    float inv  = rsqrtf((float)var + BN_EPS);
    float s    = g[c] * inv;
    scaleshift[c]      = s;
    scaleshift[64 + c] = b[c] - (float)mu * s;
}

/* ---------------- h = relu(t*scale + shift), float4 stream ---------------- */
__global__ void bn_apply_kernel(const float* __restrict__ t,
                                const float* __restrict__ ss,
                                float* __restrict__ h) {
    long i = (long)blockIdx.x * blockDim.x + threadIdx.x;  // float4 index, 3.2M exact
    float4 v = ((const float4*)t)[i];
    int c = ((int)(i & 15)) * 4;
    float a;
    a = v.x * ss[c]     + ss[64 + c];     v.x = a > 0.f ? a : 0.f;
    a = v.y * ss[c + 1] + ss[64 + c + 1]; v.y = a > 0.f ? a : 0.f;
    a = v.z * ss[c + 2] + ss[64 + c + 2]; v.z = a > 0.f ? a : 0.f;
    a = v.w * ss[c + 3] + ss[64 + c + 3]; v.w = a > 0.f ? a : 0.f;
    ((float4*)h)[i] = v;
}

/* ---------------- mean over runs + segment-sum pool ----------------------- */
__global__ void pool_mean_kernel(const float* __restrict__ h,
                                 const int* __restrict__ batch,
                                 float* __restrict__ pooled) {
    int n = blockIdx.x * 4 + (threadIdx.x >> 6);  // 5000 blocks, exact
    int c = threadIdx.x & 63;
    float s = 0.f;
#pragma unroll
    for (int r = 0; r < N_RUNS; ++r)
        s += h[((long)r * N_NODES + n) * FDIM + c];
    atomicAdd(&pooled[(long)batch[n] * FDIM + c], s * (1.f / N_RUNS));
}

/* ---------------- out += pooled @ fc_w + fc_b ----------------------------- */
__global__ void fc_accum_kernel(const float* __restrict__ pooled,
                                const float* __restrict__ w,
                                const float* __restrict__ b,
                                float* __restrict__ out) {
    int g   = blockIdx.x;      // 128
    int cls = threadIdx.x;     // 32, active < 10
    if (cls < N_CLASSES) {
        float s = b[cls];
        for (int c = 0; c < FDIM; ++c)
            s += pooled[g * FDIM + c] * w[c * N_CLASSES + cls];
        out[g * N_CLASSES + cls] += s;
    }
}

/* ---------------- row-wise log-softmax ------------------------------------ */
__global__ void log_softmax_kernel(float* out) {
    int g = threadIdx.x;  // 128
    if (g < N_GRAPHS) {
        float v[N_CLASSES], mx = -1e30f;
#pragma unroll
        for (int i = 0; i < N_CLASSES; ++i) { v[i] = out[g * N_CLASSES + i]; mx = fmaxf(mx, v[i]); }
        float s = 0.f;
#pragma unroll
        for (int i = 0; i < N_CLASSES; ++i) s += expf(v[i] - mx);
        float l = logf(s);
#pragma unroll
        for (int i = 0; i < N_CLASSES; ++i) out[g * N_CLASSES + i] = v[i] - mx - l;
    }
}

extern "C" void kernel_launch(void* const* d_in, const int* in_sizes, int n_in,
                              void* d_out, int out_size, void* d_ws, size_t ws_size,
                              hipStream_t stream) {
    (void)in_sizes; (void)n_in; (void)out_size; (void)ws_size;
    const float* x       = (const float*)d_in[0];
    const int*   edge    = (const int*)d_in[1];
    const int*   batch   = (const int*)d_in[2];
    const int*   dmask   = (const int*)d_in[3];
    const float* conv_w1 = (const float*)d_in[4];
    const float* conv_b1 = (const float*)d_in[5];
    const float* bn1_g   = (const float*)d_in[6];
    const float* bn1_b   = (const float*)d_in[7];
    const float* conv_w2 = (const float*)d_in[8];
    const float* conv_b2 = (const float*)d_in[9];
    const float* bn_g    = (const float*)d_in[10];
    const float* bn_b    = (const float*)d_in[11];
    const float* fc_w    = (const float*)d_in[12];
    const float* fc_b    = (const float*)d_in[13];
    float* out = (float*)d_out;

    const size_t BUF = (size_t)RN * FDIM * sizeof(float);  // 51,200,000 B
    char*   ws     = (char*)d_ws;
    float*  buf_h  = (float*)ws;
    float*  buf_t  = (float*)(ws + BUF);
    double* stats  = (double*)(ws + 2 * BUF);
    float*  ss     = (float*)(ws + 2 * BUF + 1024);
    float*  pooled = (float*)(ws + 2 * BUF + 1536);

    const int* src = edge;
    const int* dst = edge + N_EDGES;

    (void)hipMemsetAsync(d_out, 0, N_GRAPHS * N_CLASSES * sizeof(float), stream);
    init_h_kernel<<<RN / 4, 256, 0, stream>>>(x, dmask, buf_h);

    auto level = [&](int i) {
        (void)hipMemsetAsync(pooled, 0, N_GRAPHS * FDIM * sizeof(float), stream);
        pool_mean_kernel<<<N_NODES / 4, 256, 0, stream>>>(buf_h, batch, pooled);
        fc_accum_kernel<<<N_GRAPHS, 32, 0, stream>>>(pooled, fc_w + i * FDIM * N_CLASSES,
                                                     fc_b + i * N_CLASSES, out);
    };

    level(0);
    for (int i = 0; i < 4; ++i) {
        // agg = scatter-add(h)
        (void)hipMemsetAsync(buf_t, 0, BUF, stream);
        scatter_add_kernel<<<(N_RUNS * N_EDGES) / 4, 256, 0, stream>>>(buf_h, src, dst, buf_t);
        // t = (h+agg) @ W1 + b1   (in place over agg)
        gemm64_kernel<0><<<RN / 16 / 4, 128, 0, stream>>>(buf_h, buf_t,
            conv_w1 + i * 4096, conv_b1 + i * 64, nullptr, buf_t);
        // BN1 stats -> affine
        (void)hipMemsetAsync(stats, 0, 128 * sizeof(double), stream);
        bn_stats_kernel<<<512, 256, 0, stream>>>(buf_t, stats);
        bn_finalize_kernel<<<1, 64, 0, stream>>>(stats, bn1_g + i * 64, bn1_b + i * 64, ss);
        // t = relu(BN1(t)) @ W2 + b2   (BN+ReLU fused on load, in place)
        gemm64_kernel<1><<<RN / 16 / 4, 128, 0, stream>>>(buf_t, nullptr,
            conv_w2 + i * 4096, conv_b2 + i * 64, ss, buf_t);
        // BN2 stats -> affine, then h = relu(BN2(t))
        (void)hipMemsetAsync(stats, 0, 128 * sizeof(double), stream);
        bn_stats_kernel<<<512, 256, 0, stream>>>(buf_t, stats);
        bn_finalize_kernel<<<1, 64, 0, stream>>>(stats, bn_g + i * 64, bn_b + i * 64, ss);
        bn_apply_kernel<<<(RN * FDIM / 4) / 256, 256, 0, stream>>>(buf_t, ss, buf_h);
        level(i + 1);
    }
    log_softmax_kernel<<<1, 128, 0, stream>>>(out);
}